// ContextQueryAttn_91036126806930
// MI455X (gfx1250) — compile-verified
//
#include <hip/hip_runtime.h>
#include <hip/hip_bf16.h>

// ContextQueryAttn for MI455X (gfx1250, wave32).
// B=32, Lc=1024, Lq=128, d=128. fp32 throughout, V_WMMA_F32_16X16X4_F32 GEMMs.
// Staging: B128 global loads, TDM (tensor_load_to_lds) for the query tile.

#define NB 32
#define LC 1024
#define LQ 128
#define DD 128
#define PAD 132            // padded LDS row stride (floats): 16B aligned, bank-safe
#define NEG_INF_F (-1e30f)

typedef float v2f __attribute__((ext_vector_type(2)));
typedef float v8f __attribute__((ext_vector_type(8)));
typedef unsigned int v4u __attribute__((ext_vector_type(4)));
typedef int v4i __attribute__((ext_vector_type(4)));
typedef int v8i __attribute__((ext_vector_type(8)));

__device__ __forceinline__ v8f wmma4(v2f a, v2f b, v8f c) {
  // 8 args: (neg_a, A, neg_b, B, c_mod, C, reuse_a, reuse_b)
  return __builtin_amdgcn_wmma_f32_16x16x4_f32(false, a, false, b, (short)0, c,
                                               false, false);
}

#if __has_builtin(__builtin_amdgcn_tensor_load_to_lds) && \
    __has_builtin(__builtin_amdgcn_s_wait_tensorcnt)
#define HAVE_TDM 1
#else
#define HAVE_TDM 0
#endif

#if HAVE_TDM
// TDM: DMA a 2D f32 tile [rows x cols] (row stride = cols) from global into
// LDS with a 4-DWORD pad after every 128 DWORDs (-> LDS row stride = PAD=132).
// Caller: issue from ONE wave only; follow with s_wait_tensorcnt + barrier.
__device__ __forceinline__ void tdm_load_128x128_pad(void* lds_dst,
                                                     const float* gsrc) {
  unsigned int lds_off = (unsigned int)(unsigned long long)lds_dst; // low32 = LDS offset
  unsigned long long ga = (unsigned long long)gsrc;
  v4u g0;
  g0.x = 1u;                                 // count=1, user descriptor
  g0.y = lds_off;                            // lds_addr
  g0.z = (unsigned int)ga;                   // global_addr[31:0]
  g0.w = (unsigned int)((ga >> 32) & 0x01FFFFFFu) | (2u << 30); // addr[56:32]|type=2
  v8i g1;
  g1[0] = (int)((2u << 16)      // data_size = 4B
              | (1u << 20)      // pad_enable
              | (6u << 22)      // pad_interval: 128 DWORDs
              | (3u << 25));    // pad_amount: 4 DWORDs
  g1[1] = (int)(128u << 16);    // tensor_dim0[15:0] at bits[63:48]
  g1[2] = (int)(128u << 16);    // tensor_dim1[15:0] at bits[95:80]
  g1[3] = (int)(128u << 16);    // tile_dim0 at bits[127:112]
  g1[4] = (int)128u;            // tile_dim1 at bits[143:128]
  g1[5] = (int)128u;            // tensor_dim0_stride[31:0] at bits[191:160]
  g1[6] = 0;
  g1[7] = 0;
  v4i gz = {0, 0, 0, 0};
#if __clang_major__ >= 23
  v8i gz8 = {0, 0, 0, 0, 0, 0, 0, 0};
  __builtin_amdgcn_tensor_load_to_lds(g0, g1, gz, gz, gz8, 0);
#else
  __builtin_amdgcn_tensor_load_to_lds(g0, g1, gz, gz, 0);
#endif
}
#endif

// ---------------------------------------------------------------- K0: biases
// cb[b,i] = context[b,i,:].wc ; qb[b,j] = query[b,j,:].wq  (one wave per row)
__global__ __launch_bounds__(256) void k_bias(const float* __restrict__ ctx,
                                              const float* __restrict__ qry,
                                              const float* __restrict__ w0,
                                              float* __restrict__ cb,
                                              float* __restrict__ qb) {
  int w = threadIdx.x >> 5, lane = threadIdx.x & 31;
  long gw = (long)blockIdx.x * 8 + w;
  const float* row; const float* wv; float* dst; long di;
  if (gw < (long)NB * LC) {
    row = ctx + gw * DD; wv = w0; dst = cb; di = gw;
  } else {
    long g2 = gw - (long)NB * LC;
    if (g2 >= (long)NB * LQ) return;
    row = qry + g2 * DD; wv = w0 + DD; dst = qb; di = g2;
  }
  float s = 0.f;
#pragma unroll
  for (int c = 0; c < DD; c += 32) s += row[c + lane] * wv[c + lane];
#pragma unroll
  for (int m = 16; m >= 1; m >>= 1) s += __shfl_xor(s, m, 32);
  if (lane == 0) dst[di] = s;
}

// ------------------------------------------------- K1: sim + softmax statistics
// grid (Lc/128, B). Per workgroup: 128x128 sim tile via WMMA from LDS,
// row softmax stats (full), column softmax partials (this i-tile).
__global__ __launch_bounds__(256) void k_sim(
    const float* __restrict__ ctx, const float* __restrict__ qry,
    const unsigned char* __restrict__ cmask, const unsigned char* __restrict__ qmask,
    const float* __restrict__ w0, const float* __restrict__ cbW,
    const float* __restrict__ qbW, float* __restrict__ simW,
    float* __restrict__ rmW, float* __restrict__ rsW,
    float* __restrict__ cpmW, float* __restrict__ cpsW) {
  __shared__ float cw[128][PAD];          // (context*wm)[i0+r][k]
  __shared__ float qv[128][PAD];          // query[j][k]
  __shared__ float wstat[8][2][128];      // per-wave column (max,sumexp)

  const int b = blockIdx.y, it = blockIdx.x, i0 = it * 128;
  const float* ctxb = ctx + ((size_t)b * LC + i0) * DD;
  const float* qryb = qry + (size_t)b * LQ * DD;
  const float4* ctx4 = (const float4*)ctxb;
  const float4* wm4 = (const float4*)(w0 + 2 * DD);

#if HAVE_TDM
  if (threadIdx.x < 32) {                 // wave 0 kicks the TDM for query tile
    tdm_load_128x128_pad(&qv[0][0], qryb);
  }
#endif
  // all waves stage context*wm with B128 loads (overlaps the TDM)
  for (int idx = threadIdx.x; idx < 128 * 32; idx += 256) {
    int r = idx >> 5, c4 = idx & 31;
    float4 v = ctx4[idx];
    float4 wv = wm4[c4];
    float4 o;
    o.x = v.x * wv.x; o.y = v.y * wv.y; o.z = v.z * wv.z; o.w = v.w * wv.w;
    *(float4*)&cw[r][c4 * 4] = o;
  }
#if HAVE_TDM
  if (threadIdx.x < 32) __builtin_amdgcn_s_wait_tensorcnt(0);
#else
  {
    const float4* qry4 = (const float4*)qryb;
    for (int idx = threadIdx.x; idx < 128 * 32; idx += 256) {
      int r = idx >> 5, c4 = idx & 31;
      *(float4*)&qv[r][c4 * 4] = qry4[idx];
    }
  }
#endif
  __syncthreads();

  const int w = threadIdx.x >> 5, lane = threadIdx.x & 31;
  const int half = lane >> 4, lm = lane & 15, koff = half * 2;
  const int rowm = 16 * w;

  v8f acc[8];
#pragma unroll
  for (int t = 0; t < 8; ++t) {
    int n0 = t * 16;
    v8f c = {};
    for (int k = 0; k < DD; k += 4) {
      v2f a, bb;
      a.x = cw[rowm + lm][k + koff];
      a.y = cw[rowm + lm][k + koff + 1];
      bb.x = qv[n0 + lm][k + koff];
      bb.y = qv[n0 + lm][k + koff + 1];
      c = wmma4(a, bb, c);
    }
    acc[t] = c;
  }

  // biases + mask + store sim
  float cbv[8]; unsigned char cmv[8];
#pragma unroll
  for (int v = 0; v < 8; ++v) {
    int gi = i0 + rowm + v + half * 8;
    cbv[v] = cbW[b * LC + gi];
    cmv[v] = cmask[b * LC + gi];
  }
#pragma unroll
  for (int t = 0; t < 8; ++t) {
    int j = t * 16 + lm;
    float qbv = qbW[b * LQ + j];
    unsigned char qmv = qmask[b * LQ + j];
#pragma unroll
    for (int v = 0; v < 8; ++v) {
      float s = acc[t][v] + cbv[v] + qbv;
      if (cmv[v] | qmv) s = NEG_INF_F;
      acc[t][v] = s;
      simW[((size_t)b * LC + i0 + rowm + v + half * 8) * LQ + j] = s;
    }
  }

  // row softmax stats: rows live at (v, half); values spread over lm & tiles
#pragma unroll
  for (int v = 0; v < 8; ++v) {
    float pm = acc[0][v];
#pragma unroll
    for (int t = 1; t < 8; ++t) pm = fmaxf(pm, acc[t][v]);
#pragma unroll
    for (int m = 8; m >= 1; m >>= 1) pm = fmaxf(pm, __shfl_xor(pm, m, 32));
    float ps = 0.f;
#pragma unroll
    for (int t = 0; t < 8; ++t) ps += __expf(acc[t][v] - pm);
#pragma unroll
    for (int m = 8; m >= 1; m >>= 1) ps += __shfl_xor(ps, m, 32);
    if (lm == 0) {
      int gi = i0 + rowm + v + half * 8;
      rmW[b * LC + gi] = pm;
      rsW[b * LC + gi] = ps;
    }
  }

  // column partials over this wave's 16 rows
#pragma unroll
  for (int t = 0; t < 8; ++t) {
    float cml = acc[t][0];
#pragma unroll
    for (int v = 1; v < 8; ++v) cml = fmaxf(cml, acc[t][v]);
    float cmx = fmaxf(cml, __shfl_xor(cml, 16, 32));
    float csl = 0.f;
#pragma unroll
    for (int v = 0; v < 8; ++v) csl += __expf(acc[t][v] - cmx);
    float cs2 = csl + __shfl_xor(csl, 16, 32);
    if (half == 0) {
      wstat[w][0][t * 16 + lm] = cmx;
      wstat[w][1][t * 16 + lm] = cs2;
    }
  }
  __syncthreads();

  if (threadIdx.x < 128) {
    int j = threadIdx.x;
    float M = wstat[0][0][j];
#pragma unroll
    for (int ww = 1; ww < 8; ++ww) M = fmaxf(M, wstat[ww][0][j]);
    float S = 0.f;
#pragma unroll
    for (int ww = 0; ww < 8; ++ww) S += wstat[ww][1][j] * __expf(wstat[ww][0][j] - M);
    cpmW[((size_t)b * 8 + it) * LQ + j] = M;
    cpsW[((size_t)b * 8 + it) * LQ + j] = S;
  }
}

// ----------------------------------------- K2: merge column partials (8 tiles)
__global__ __launch_bounds__(256) void k_colmerge(const float* __restrict__ cpmW,
                                                  const float* __restrict__ cpsW,
                                                  float* __restrict__ cmW,
                                                  float* __restrict__ csW) {
  int idx = blockIdx.x * 256 + threadIdx.x;
  if (idx >= NB * LQ) return;
  int b = idx / LQ, j = idx % LQ;
  float M = cpmW[((size_t)b * 8) * LQ + j];
#pragma unroll
  for (int t = 1; t < 8; ++t) M = fmaxf(M, cpmW[((size_t)b * 8 + t) * LQ + j]);
  float S = 0.f;
#pragma unroll
  for (int t = 0; t < 8; ++t)
    S += cpsW[((size_t)b * 8 + t) * LQ + j] * __expf(cpmW[((size_t)b * 8 + t) * LQ + j] - M);
  cmW[idx] = M;
  csW[idx] = S;
}

// -------------------------------------------------- K3: C2 = SS^T @ context
// grid (2, B): j-half per workgroup, K=1024 staged in 128-tiles.
__global__ __launch_bounds__(256) void k_c2(const float* __restrict__ ctx,
                                            const float* __restrict__ simW,
                                            const float* __restrict__ cmW,
                                            const float* __restrict__ csW,
                                            float* __restrict__ c2W) {
  __shared__ float aLDS[64][PAD];  // exp(sim^T) for current k-tile
  const int b = blockIdx.y, j0 = blockIdx.x * 64;
  const int w = threadIdx.x >> 5, lane = threadIdx.x & 31;
  const int half = lane >> 4, lm = lane & 15, koff = half * 2;
  const int jr = 16 * (w & 3);     // row strip within the 64 j's
  const int nh = w >> 2;           // column half (dd)
  const float* ctxb = ctx + (size_t)b * LC * DD;
  const float4* sim4 = (const float4*)simW;

  v8f acc[4] = {};
  for (int k0 = 0; k0 < LC; k0 += 128) {
    for (int idx = threadIdx.x; idx < 128 * 16; idx += 256) {
      int kk = idx >> 4, jj4 = idx & 15;
      float4 x = sim4[((size_t)b * LC + k0 + kk) * (LQ / 4) + (j0 >> 2) + jj4];
      float4 cm = *(const float4*)&cmW[b * LQ + j0 + jj4 * 4];
      aLDS[jj4 * 4 + 0][kk] = __expf(x.x - cm.x);
      aLDS[jj4 * 4 + 1][kk] = __expf(x.y - cm.y);
      aLDS[jj4 * 4 + 2][kk] = __expf(x.z - cm.z);
      aLDS[jj4 * 4 + 3][kk] = __expf(x.w - cm.w);
    }
    __syncthreads();
    for (int k = 0; k < 128; k += 4) {
      v2f a;
      a.x = aLDS[jr + lm][k + koff];
      a.y = aLDS[jr + lm][k + koff + 1];
      size_t kg = (size_t)(k0 + k + koff);
#pragma unroll
      for (int tt = 0; tt < 4; ++tt) {
        int n = nh * 64 + tt * 16 + lm;
        v2f bb;
        bb.x = ctxb[kg * DD + n];
        bb.y = ctxb[(kg + 1) * DD + n];
        acc[tt] = wmma4(a, bb, acc[tt]);
      }
    }
    __syncthreads();
  }
#pragma unroll
  for (int tt = 0; tt < 4; ++tt) {
#pragma unroll
    for (int v = 0; v < 8; ++v) {
      int j = j0 + jr + v + half * 8;
      int n = nh * 64 + tt * 16 + lm;
      c2W[((size_t)b * LQ + j) * DD + n] = acc[tt][v] / csW[b * LQ + j];
    }
  }
}

// --------------------------------------- K4: A = S @ query ; Bout = S @ C2
// grid (Lc/128, B): stage normalized S strip once, two WMMA passes.
__global__ __launch_bounds__(256) void k_out(const float* __restrict__ qry,
                                             const float* __restrict__ simW,
                                             const float* __restrict__ rmW,
                                             const float* __restrict__ rsW,
                                             const float* __restrict__ c2W,
                                             float* __restrict__ outA,
                                             float* __restrict__ outB) {
  __shared__ float sLDS[128][PAD];
  const int b = blockIdx.y, i0 = blockIdx.x * 128;
  const float4* sim4 = (const float4*)simW;

  for (int idx = threadIdx.x; idx < 128 * 32; idx += 256) {
    int r = idx >> 5, j4 = idx & 31;
    float4 x = sim4[((size_t)b * LC + i0 + r) * (LQ / 4) + j4];
    float m = rmW[b * LC + i0 + r];
    float inv = 1.0f / rsW[b * LC + i0 + r];
    float4 o;
    o.x = __expf(x.x - m) * inv; o.y = __expf(x.y - m) * inv;
    o.z = __expf(x.z - m) * inv; o.w = __expf(x.w - m) * inv;
    *(float4*)&sLDS[r][j4 * 4] = o;
  }
  __syncthreads();

  const int w = threadIdx.x >> 5, lane = threadIdx.x & 31;
  const int half = lane >> 4, lm = lane & 15, koff = half * 2;
  const int rowm = 16 * w;

  for (int pass = 0; pass < 2; ++pass) {
    const float* Bmat = (pass == 0) ? (qry + (size_t)b * LQ * DD)
                                    : (c2W + (size_t)b * LQ * DD);
    float* outp = (pass == 0) ? outA : outB;
#pragma unroll
    for (int t = 0; t < 8; ++t) {
      int n0 = t * 16;
      v8f c = {};
      for (int k = 0; k < LQ; k += 4) {
        v2f a, bb;
        a.x = sLDS[rowm + lm][k + koff];
        a.y = sLDS[rowm + lm][k + koff + 1];
        bb.x = Bmat[(size_t)(k + koff) * DD + n0 + lm];
        bb.y = Bmat[(size_t)(k + koff + 1) * DD + n0 + lm];
        c = wmma4(a, bb, c);
      }
#pragma unroll
      for (int v = 0; v < 8; ++v)
        outp[((size_t)b * LC + i0 + rowm + v + half * 8) * DD + n0 + lm] = c[v];
    }
  }
}

extern "C" void kernel_launch(void* const* d_in, const int* in_sizes, int n_in,
                              void* d_out, int out_size, void* d_ws, size_t ws_size,
                              hipStream_t stream) {
  const float* ctx = (const float*)d_in[0];
  const float* qry = (const float*)d_in[1];
  const unsigned char* cmask = (const unsigned char*)d_in[2];
  const unsigned char* qmask = (const unsigned char*)d_in[3];
  const float* w0 = (const float*)d_in[4];
  float* out = (float*)d_out;

  float* ws = (float*)d_ws;
  const size_t NSIM = (size_t)NB * LC * LQ;
  float* simW = ws;
  float* cbW  = simW + NSIM;
  float* qbW  = cbW + (size_t)NB * LC;
  float* rmW  = qbW + (size_t)NB * LQ;
  float* rsW  = rmW + (size_t)NB * LC;
  float* cpmW = rsW + (size_t)NB * LC;
  float* cpsW = cpmW + (size_t)NB * 8 * LQ;
  float* cmW  = cpsW + (size_t)NB * 8 * LQ;
  float* csW  = cmW + (size_t)NB * LQ;
  float* c2W  = csW + (size_t)NB * LQ;

  // K0: biases (36864 rows, 8 waves/block)
  k_bias<<<dim3((NB * LC + NB * LQ) / 8), dim3(256), 0, stream>>>(ctx, qry, w0, cbW, qbW);
  // K1: sim + stats
  k_sim<<<dim3(LC / 128, NB), dim3(256), 0, stream>>>(ctx, qry, cmask, qmask, w0,
                                                      cbW, qbW, simW, rmW, rsW,
                                                      cpmW, cpsW);
  // K2: column stat merge
  k_colmerge<<<dim3((NB * LQ + 255) / 256), dim3(256), 0, stream>>>(cpmW, cpsW, cmW, csW);
  // K3: C2 = SS^T @ context
  k_c2<<<dim3(2, NB), dim3(256), 0, stream>>>(ctx, simW, cmW, csW, c2W);
  // K4: outputs
  k_out<<<dim3(LC / 128, NB), dim3(256), 0, stream>>>(qry, simW, rmW, rsW, c2W,
                                                      out, out + (size_t)NB * LC * DD);
}